// PointNetSetAbstraction_85332410237538
// MI455X (gfx1250) — compile-verified
//
#include <hip/hip_runtime.h>
#include <hip/hip_bf16.h>

// PointNet++ Set Abstraction for MI455X (gfx1250), wave32 + WMMA f32 16x16x4.
//
// Pipeline: FPS (LDS-resident, sequential argmax) -> kNN top-32 (LDS iterative
// argmin) -> gather/pad -> 3x [WMMA GEMM + deterministic global-BN stats]
// -> fused affine/ReLU + max-over-K pooling.
//
// All reductions use fixed-order trees (no float atomics) so every call
// produces bit-identical output.

#define B_  16
#define N_  4096
#define D_  6
#define S_  1024   // NUM_CENTERS
#define K_  32
#define EPS_ 1e-5f

static constexpr int ROWS = B_ * S_ * K_;   // 524288 rows through the MLP
static constexpr int NBLK = ROWS / 128;     // 4096 GEMM blocks (8 waves x 16 rows)

typedef __attribute__((ext_vector_type(2))) float v2f;
typedef __attribute__((ext_vector_type(8))) float v8f;

// ---------------------------------------------------------------- 1) FPS
// One block per batch. Points SoA in LDS; each of 1024 threads owns 4 points'
// running min-distance in registers. Per step: register argmax -> wave
// shfl_xor reduce -> 32-entry LDS reduce -> broadcast new center.
__global__ __launch_bounds__(1024) void fps_kernel(const float* __restrict__ xyz,
                                                   float* __restrict__ out_centers_xyz) {
    __shared__ float lx[N_], ly[N_], lz[N_];
    __shared__ float wbest[32];
    __shared__ int   wbidx[32];
    __shared__ float cpt[3];

    const int b = blockIdx.x;
    const int t = threadIdx.x;
    const float* xb = xyz + (size_t)b * N_ * 3;
    for (int i = t; i < N_; i += 1024) {
        lx[i] = xb[i * 3 + 0];
        ly[i] = xb[i * 3 + 1];
        lz[i] = xb[i * 3 + 2];
    }
    __syncthreads();

    const float c0x = lx[0], c0y = ly[0], c0z = lz[0];
    float dmin[4];
#pragma unroll
    for (int i = 0; i < 4; ++i) {
        const int p = t + i * 1024;
        const float dx = lx[p] - c0x, dy = ly[p] - c0y, dz = lz[p] - c0z;
        dmin[i] = dx * dx + dy * dy + dz * dz;
    }
    if (t == 0) {
        out_centers_xyz[(size_t)b * S_ * 3 + 0] = c0x;
        out_centers_xyz[(size_t)b * S_ * 3 + 1] = c0y;
        out_centers_xyz[(size_t)b * S_ * 3 + 2] = c0z;
    }

    const int lane = t & 31;
    const int wave = t >> 5;
    for (int s = 1; s < S_; ++s) {
        // local argmax over this thread's 4 points (ties -> lower index)
        float bd = dmin[0];
        int   bp = t;
#pragma unroll
        for (int i = 1; i < 4; ++i) {
            const int p = t + i * 1024;
            if (dmin[i] > bd || (dmin[i] == bd && p < bp)) { bd = dmin[i]; bp = p; }
        }
        // wave32 reduce
#pragma unroll
        for (int off = 16; off > 0; off >>= 1) {
            const float od = __shfl_xor(bd, off, 32);
            const int   op = __shfl_xor(bp, off, 32);
            if (od > bd || (od == bd && op < bp)) { bd = od; bp = op; }
        }
        if (lane == 0) { wbest[wave] = bd; wbidx[wave] = bp; }
        __syncthreads();
        if (t == 0) {
            float fb = wbest[0];
            int   fp = wbidx[0];
            for (int w = 1; w < 32; ++w)
                if (wbest[w] > fb || (wbest[w] == fb && wbidx[w] < fp)) { fb = wbest[w]; fp = wbidx[w]; }
            cpt[0] = lx[fp]; cpt[1] = ly[fp]; cpt[2] = lz[fp];
            out_centers_xyz[((size_t)b * S_ + s) * 3 + 0] = lx[fp];
            out_centers_xyz[((size_t)b * S_ + s) * 3 + 1] = ly[fp];
            out_centers_xyz[((size_t)b * S_ + s) * 3 + 2] = lz[fp];
        }
        __syncthreads();
        const float cx = cpt[0], cy = cpt[1], cz = cpt[2];
#pragma unroll
        for (int i = 0; i < 4; ++i) {
            const int p = t + i * 1024;
            const float dx = lx[p] - cx, dy = ly[p] - cy, dz = lz[p] - cz;
            const float nd = dx * dx + dy * dy + dz * dz;
            dmin[i] = fminf(dmin[i], nd);
        }
        // cpt is protected by the __syncthreads() after wbest writes next iter
    }
}

// ---------------------------------------------------------------- 2) kNN
// One block (256 threads) per center. Distances in LDS; 32 iterative argmin
// passes (only the *set* of 32 nearest matters downstream).
__global__ __launch_bounds__(256) void knn_kernel(const float* __restrict__ xyz,
                                                  const float* __restrict__ centers_xyz,
                                                  int* __restrict__ group_idx) {
    __shared__ float dist[N_];
    __shared__ float wmin[8];
    __shared__ int   wmidx[8];

    const int g = blockIdx.x;          // b*S + s
    const int b = g >> 10;
    const int t = threadIdx.x;
    const float cx = centers_xyz[(size_t)g * 3 + 0];
    const float cy = centers_xyz[(size_t)g * 3 + 1];
    const float cz = centers_xyz[(size_t)g * 3 + 2];
    const float cc = cx * cx + cy * cy + cz * cz;
    const float* xb = xyz + (size_t)b * N_ * 3;

    for (int i = t; i < N_; i += 256) {
        const float px = xb[i * 3 + 0], py = xb[i * 3 + 1], pz = xb[i * 3 + 2];
        // reference formula: |c|^2 + |p|^2 - 2 c.p, clamped at 0
        const float d = cc + (px * px + py * py + pz * pz) - 2.0f * (cx * px + cy * py + cz * pz);
        dist[i] = fmaxf(d, 0.0f);
    }
    __syncthreads();

    const int lane = t & 31;
    const int wave = t >> 5;
    for (int k = 0; k < K_; ++k) {
        float bd = 1e30f;
        int   bi = N_;
        for (int i = t; i < N_; i += 256) {
            const float d = dist[i];
            if (d < bd || (d == bd && i < bi)) { bd = d; bi = i; }
        }
#pragma unroll
        for (int off = 16; off > 0; off >>= 1) {
            const float od = __shfl_xor(bd, off, 32);
            const int   oi = __shfl_xor(bi, off, 32);
            if (od < bd || (od == bd && oi < bi)) { bd = od; bi = oi; }
        }
        if (lane == 0) { wmin[wave] = bd; wmidx[wave] = bi; }
        __syncthreads();
        if (t == 0) {
            float fb = wmin[0];
            int   fi = wmidx[0];
            for (int w = 1; w < 8; ++w)
                if (wmin[w] < fb || (wmin[w] == fb && wmidx[w] < fi)) { fb = wmin[w]; fi = wmidx[w]; }
            group_idx[(size_t)g * K_ + k] = fi;
            dist[fi] = 1e30f;
        }
        __syncthreads();
    }
}

// ---------------------------------------------------------------- 3) gather
// Build padded A-matrix rows: [dx,dy,dz, f0..f5, 0,0,0]  (K padded 9 -> 12)
__global__ __launch_bounds__(256) void gather_kernel(const float* __restrict__ xyz,
                                                     const float* __restrict__ feat,
                                                     const float* __restrict__ centers_xyz,
                                                     const int* __restrict__ group_idx,
                                                     float* __restrict__ Apad) {
    const int row = blockIdx.x * 256 + threadIdx.x;
    if (row >= ROWS) return;
    const int j = row & (K_ - 1);
    const int g = row / K_;            // b*S + s
    const int b = g >> 10;
    const int p = group_idx[(size_t)g * K_ + j];
    const float* xp = xyz + ((size_t)b * N_ + p) * 3;
    const float* cp = centers_xyz + (size_t)g * 3;
    const float* fp = feat + ((size_t)b * N_ + p) * D_;
    float* a = Apad + (size_t)row * 12;
    a[0] = xp[0] - cp[0];
    a[1] = xp[1] - cp[1];
    a[2] = xp[2] - cp[2];
#pragma unroll
    for (int c = 0; c < 6; ++c) a[3 + c] = fp[c];
    a[9] = 0.0f; a[10] = 0.0f; a[11] = 0.0f;
}

// ---------------------------------------------------------------- GEMM + BN stats
// X[ROWS x NDIM] = act(A[ROWS x KDIM]) @ W[KSRC x NDIM]  via V_WMMA_F32_16X16X4_F32.
// 256 threads = 8 waves; each wave owns a 16-row strip across all NDIM columns.
// W is staged in LDS pre-paired as float2 fragments (Bs2[k/2][n] = {W[k][n],
// W[k+1][n]}) so each B fragment is a single aligned ds_load_b64.
// AFF: apply y = relu(a*scale[k] + shift[k]) to A elements on load (fused BN+ReLU
// of the previous layer).
// Per-channel sum/sumsq: each (wave, half-wave) writes its private partial to a
// unique LDS slot; a fixed-order per-block reduction writes part[block][ch]
// (fully deterministic, no atomics).
template <int KDIM, int KSRC, int NDIM, bool AFF>
__global__ __launch_bounds__(256) void gemm_bn_kernel(const float* __restrict__ A,
                                                      const float* __restrict__ W,
                                                      const float* __restrict__ scale,
                                                      const float* __restrict__ shift,
                                                      float* __restrict__ X,
                                                      float* __restrict__ part_sum,
                                                      float* __restrict__ part_sq) {
    __shared__ v2f   Bs2[(KDIM / 2) * NDIM];
    __shared__ float psum[NDIM * 16];   // [col][wave*2 + hi]
    __shared__ float psq[NDIM * 16];
    __shared__ float ssc[KDIM];
    __shared__ float ssh[KDIM];

    const int t = threadIdx.x;
    float* BsRaw = (float*)Bs2;
    for (int i = t; i < KDIM * NDIM; i += 256) {
        const int kr = i / NDIM;
        const int n  = i - kr * NDIM;
        const float val = (kr < KSRC) ? W[(size_t)kr * NDIM + n] : 0.0f;
        BsRaw[((kr >> 1) * NDIM + n) * 2 + (kr & 1)] = val;
    }
    if (AFF) {
        for (int i = t; i < KDIM; i += 256) { ssc[i] = scale[i]; ssh[i] = shift[i]; }
    }
    __syncthreads();

    constexpr int NT = NDIM / 16;
    v8f C[NT] = {};
    const int lane = t & 31;
    const int wave = t >> 5;
    const int m0   = (blockIdx.x * 8 + wave) * 16;
    const int mr   = lane & 15;          // A row within tile
    const int khalf = lane >> 4;         // 0 or 1: which K pair this lane holds
    const int koff  = khalf << 1;        // 0 or 2
    const int nsel  = lane & 15;         // B/D column within 16-wide tile

    for (int k0 = 0; k0 < KDIM; k0 += 4) {
        const float* ap = A + (size_t)(m0 + mr) * KDIM + (k0 + koff);
        v2f a = *(const v2f*)ap;
        if (AFF) {
            a.x = fmaxf(fmaf(a.x, ssc[k0 + koff],     ssh[k0 + koff]),     0.0f);
            a.y = fmaxf(fmaf(a.y, ssc[k0 + koff + 1], ssh[k0 + koff + 1]), 0.0f);
        }
        const v2f* brow = Bs2 + ((k0 >> 1) + khalf) * NDIM + nsel;
#pragma unroll
        for (int nt = 0; nt < NT; ++nt) {
            v2f bf = brow[nt * 16];
            C[nt] = __builtin_amdgcn_wmma_f32_16x16x4_f32(
                false, a, false, bf, (short)0, C[nt], false, false);
        }
    }

    // store pre-BN activations + deterministic per-block channel stats
    const int rbase = khalf * 8;         // lanes 16-31 hold rows m0+8..m0+15
    const int slot  = wave * 2 + khalf;  // unique writer per (col, slot)
#pragma unroll
    for (int nt = 0; nt < NT; ++nt) {
        const int col = nt * 16 + nsel;
        float s = 0.0f, q = 0.0f;
#pragma unroll
        for (int r = 0; r < 8; ++r) {
            const float v = C[nt][r];
            X[(size_t)(m0 + rbase + r) * NDIM + col] = v;
            s += v;
            q += v * v;
        }
        psum[col * 16 + slot] = s;
        psq[col * 16 + slot]  = q;
    }
    __syncthreads();
    for (int c = t; c < NDIM; c += 256) {
        float s = 0.0f, q = 0.0f;
#pragma unroll
        for (int i = 0; i < 16; ++i) {   // fixed order -> deterministic
            s += psum[c * 16 + i];
            q += psq[c * 16 + i];
        }
        part_sum[(size_t)blockIdx.x * NDIM + c] = s;
        part_sq[(size_t)blockIdx.x * NDIM + c]  = q;
    }
}

// ---------------------------------------------------------------- BN finalize
// One block per channel; fixed-assignment strided accumulation + fixed-pattern
// LDS tree => deterministic. scale = g*rsqrt(var+eps), shift = b - mean*scale.
__global__ __launch_bounds__(256) void finalize_stats(const float* __restrict__ part_sum,
                                                      const float* __restrict__ part_sq,
                                                      const float* __restrict__ g,
                                                      const float* __restrict__ bb,
                                                      float* __restrict__ scale,
                                                      float* __restrict__ shift,
                                                      int C, int nblk) {
    __shared__ float rsum[256];
    __shared__ float rsq[256];
    const int c = blockIdx.x;
    const int t = threadIdx.x;
    float s = 0.0f, q = 0.0f;
    for (int i = t; i < nblk; i += 256) {
        s += part_sum[(size_t)i * C + c];
        q += part_sq[(size_t)i * C + c];
    }
    rsum[t] = s;
    rsq[t]  = q;
    __syncthreads();
    for (int off = 128; off > 0; off >>= 1) {
        if (t < off) { rsum[t] += rsum[t + off]; rsq[t] += rsq[t + off]; }
        __syncthreads();
    }
    if (t == 0) {
        const float invN = 1.0f / (float)ROWS;
        const float m  = rsum[0] * invN;
        const float v  = rsq[0] * invN - m * m;
        const float rs = rsqrtf(v + EPS_);
        const float sc = g[c] * rs;
        scale[c] = sc;
        shift[c] = bb[c] - m * sc;
    }
}

// ---------------------------------------------------------------- max pool
// Apply layer-3 affine+ReLU on the fly and max over K=32 group members.
__global__ __launch_bounds__(128) void maxpool_kernel(const float* __restrict__ X3,
                                                      const float* __restrict__ scale,
                                                      const float* __restrict__ shift,
                                                      float* __restrict__ out_feat) {
    const int g = blockIdx.x;     // b*S + s
    const int c = threadIdx.x;    // channel 0..127
    const float sc = scale[c], sh = shift[c];
    const float* p = X3 + (size_t)g * K_ * 128 + c;
    float m = -1e30f;
#pragma unroll
    for (int j = 0; j < K_; ++j)
        m = fmaxf(m, fmaxf(fmaf(p[j * 128], sc, sh), 0.0f));
    out_feat[(size_t)g * 128 + c] = m;
}

// ---------------------------------------------------------------- launch
extern "C" void kernel_launch(void* const* d_in, const int* in_sizes, int n_in,
                              void* d_out, int out_size, void* d_ws, size_t ws_size,
                              hipStream_t stream) {
    const float* xyz  = (const float*)d_in[0];
    const float* feat = (const float*)d_in[1];
    const float* W1 = (const float*)d_in[2];
    const float* g1 = (const float*)d_in[3];
    const float* b1 = (const float*)d_in[4];
    const float* W2 = (const float*)d_in[5];
    const float* g2 = (const float*)d_in[6];
    const float* b2 = (const float*)d_in[7];
    const float* W3 = (const float*)d_in[8];
    const float* g3 = (const float*)d_in[9];
    const float* b3 = (const float*)d_in[10];

    float* out_centers_xyz = (float*)d_out;                       // 16*1024*3
    float* out_feat = (float*)d_out + (size_t)B_ * S_ * 3;        // 16*1024*128

    char* w = (char*)d_ws;
    int*   group_idx = (int*)w;                                   // 2 MB
    float* stat = (float*)(w + (2 << 20));                        // scale/shift
    float* sc1 = stat +   0; float* sh1 = stat +  64;
    float* sc2 = stat + 128; float* sh2 = stat + 192;
    float* sc3 = stat + 256; float* sh3 = stat + 384;             // 512 floats

    float* part_sum = (float*)(w + (4 << 20));                    // NBLK*128 f32 (2 MB)
    float* part_sq  = (float*)(w + (6 << 20));                    // NBLK*128 f32 (2 MB)

    float* Apad = (float*)(w + (8 << 20));                        // ROWS*12 f32
    float* X2   = Apad + (size_t)ROWS * 12;                       // ROWS*64  f32
    float* X13  = X2   + (size_t)ROWS * 64;                       // ROWS*128 f32
    float* X1   = X13;   // x1 occupies the first half of X13; dead before x3 written
    float* X3   = X13;

    fps_kernel<<<B_, 1024, 0, stream>>>(xyz, out_centers_xyz);
    knn_kernel<<<B_ * S_, 256, 0, stream>>>(xyz, out_centers_xyz, group_idx);
    gather_kernel<<<ROWS / 256, 256, 0, stream>>>(xyz, feat, out_centers_xyz, group_idx, Apad);

    gemm_bn_kernel<12, 9, 64, false><<<NBLK, 256, 0, stream>>>(
        Apad, W1, nullptr, nullptr, X1, part_sum, part_sq);
    finalize_stats<<<64, 256, 0, stream>>>(part_sum, part_sq, g1, b1, sc1, sh1, 64, NBLK);

    gemm_bn_kernel<64, 64, 64, true><<<NBLK, 256, 0, stream>>>(
        X1, W2, sc1, sh1, X2, part_sum, part_sq);
    finalize_stats<<<64, 256, 0, stream>>>(part_sum, part_sq, g2, b2, sc2, sh2, 64, NBLK);

    gemm_bn_kernel<64, 64, 128, true><<<NBLK, 256, 0, stream>>>(
        X2, W3, sc2, sh2, X3, part_sum, part_sq);
    finalize_stats<<<128, 256, 0, stream>>>(part_sum, part_sq, g3, b3, sc3, sh3, 128, NBLK);

    maxpool_kernel<<<B_ * S_, 128, 0, stream>>>(X3, sc3, sh3, out_feat);
}